// SubjectInvariantLoss_10196252360791
// MI455X (gfx1250) — compile-verified
//
#include <hip/hip_runtime.h>
#include <math.h>

typedef __attribute__((ext_vector_type(2))) float v2f;
typedef __attribute__((ext_vector_type(8))) float v8f;

// ---------------------------------------------------------------------------
// ws layout (floats): sq[B] | invn[B] | num[B] | den[B] | cnt[B]
//                     | meanS[D] | meanT[D] | S[2*D*D] | scalars[16]
// scalars: 0=xx_sum 1=yy_sum 2=cross_sum 3=coral_sumsq 4=loss_sum 5=valid_cnt
// ---------------------------------------------------------------------------

__global__ void sil_init(float* ws, int n) {
  int i = blockIdx.x * blockDim.x + threadIdx.x;
  int stride = gridDim.x * blockDim.x;
  for (; i < n; i += stride) ws[i] = 0.0f;
}

// One wave per row: squared norm and inverse clamped norm.
__global__ void sil_row_norms(const float* __restrict__ F, float* __restrict__ sq,
                              float* __restrict__ invn, int D) {
  const int lane = threadIdx.x & 31;
  const int row  = blockIdx.x * (blockDim.x >> 5) + (threadIdx.x >> 5);
  const float* fr = F + (size_t)row * D;
  float s = 0.0f;
  for (int c = lane; c < D; c += 32) { float v = fr[c]; s += v * v; }
  #pragma unroll
  for (int o = 1; o < 32; o <<= 1) s += __shfl_xor(s, o, 32);
  if (lane == 0) {
    sq[row]   = s;
    invn[row] = 1.0f / fmaxf(sqrtf(s), 1e-12f);
  }
}

// One block per column: mean over src half and tgt half.
__global__ void sil_col_means(const float* __restrict__ F, float* __restrict__ meanS,
                              float* __restrict__ meanT, int B, int D) {
  __shared__ float sm[256];
  const int c = blockIdx.x, t = threadIdx.x, mid = B >> 1;
  float s0 = 0.0f, s1 = 0.0f;
  for (int r = t; r < mid; r += blockDim.x) {
    s0 += F[(size_t)r * D + c];
    s1 += F[(size_t)(r + mid) * D + c];
  }
  sm[t] = s0; __syncthreads();
  for (int o = blockDim.x >> 1; o > 0; o >>= 1) { if (t < o) sm[t] += sm[t + o]; __syncthreads(); }
  if (t == 0) meanS[c] = sm[0] / (float)mid;
  __syncthreads();
  sm[t] = s1; __syncthreads();
  for (int o = blockDim.x >> 1; o > 0; o >>= 1) { if (t < o) sm[t] += sm[t + o]; __syncthreads(); }
  if (t == 0) meanT[c] = sm[0] / (float)mid;
}

// Fused symmetric pass: one wave per UPPER-TRIANGLE 16x16 tile of G = F*F^T.
// K = D = 256 -> 64 v_wmma_f32_16x16x4_f32 per tile. G, exp(sim), and all masks
// are symmetric, so an off-diagonal tile contributes:
//   - to rows i0..i0+15 via butterfly reduction across the 16 N-lanes,
//   - to rows j0..j0+15 (the mirrored block) via per-lane column sums,
//   - MMD partials with weight 2 (both ordered directions).
__global__ void sil_gram_pass(const float* __restrict__ F,
                              const int* __restrict__ lab,
                              const int* __restrict__ sub,
                              const float* __restrict__ sq,
                              const float* __restrict__ invn,
                              float* __restrict__ rowNum,
                              float* __restrict__ rowDen,
                              float* __restrict__ rowCnt,
                              float* __restrict__ scalars,
                              int B, int D, long long totalTiles) {
  const int lane = threadIdx.x & 31;
  const int wave = threadIdx.x >> 5;
  const long long tileId = (long long)blockIdx.x * (blockDim.x >> 5) + wave;
  if (tileId >= totalTiles) return;   // wave-uniform exit (EXEC stays all-ones)

  // Decode triangular index: row ti has (T - ti) tiles, offset(t) = t*T - t(t-1)/2.
  const int T = B >> 4;
  const double dT = (double)T;
  const double disc = (2.0 * dT + 1.0) * (2.0 * dT + 1.0) - 8.0 * (double)tileId;
  int t = (int)(((2.0 * dT + 1.0) - sqrt(disc)) * 0.5);
  if (t < 0) t = 0;
  if (t > T - 1) t = T - 1;
  while (t > 0 && (long long)t * T - (long long)t * (t - 1) / 2 > tileId) --t;
  while ((long long)(t + 1) * T - (long long)(t + 1) * t / 2 <= tileId) ++t;
  const int ti = t;
  const int tj = ti + (int)(tileId - ((long long)ti * T - (long long)ti * (ti - 1) / 2));
  const bool offDiag = (ti != tj);

  const int i0 = ti << 4;
  const int j0 = tj << 4;
  const int m  = lane & 15;   // M (A) / N (B,C,D) index within tile
  const int kh = lane >> 4;   // K half: 0 -> K=k,k+1 ; 1 -> K=k+2,k+3

  // Identical fragment layout for A and B of a Gram matrix: contiguous v2f loads.
  const float* arow = F + (size_t)(i0 + m) * D + 2 * kh;
  const float* brow = F + (size_t)(j0 + m) * D + 2 * kh;

  v8f acc = {};
  for (int k = 0; k < D; k += 4) {
    v2f a = *(const v2f*)(arow + k);
    v2f b = *(const v2f*)(brow + k);
    acc = __builtin_amdgcn_wmma_f32_16x16x4_f32(false, a, false, b,
                                                (short)0, acc, false, false);
  }

  const int   mid = B >> 1;
  const int   j   = j0 + m;           // column index for this lane
  const float nj  = sq[j];
  const float inj = invn[j];
  const int   lj  = lab[j];
  const int   sj  = sub[j];
  const bool  jSrc = (j < mid);

  float xx = 0.0f, yy = 0.0f, cr = 0.0f;
  float cden = 0.0f, cnum = 0.0f, ccnt = 0.0f;   // column-side (mirrored rows j)

  #pragma unroll
  for (int r = 0; r < 8; ++r) {
    const int   i = i0 + 8 * kh + r;  // C/D layout: M = r + 8*(lane>=16)
    const float g = acc[r];

    // MMD (bandwidth 1): exp(-dist2/2)
    const float d2 = sq[i] + nj - 2.0f * g;
    const float w  = __expf(-0.5f * d2);
    const bool  iSrc = (i < mid);
    xx += (iSrc && jSrc)   ? w : 0.0f;
    yy += (!iSrc && !jSrc) ? w : 0.0f;
    cr += (iSrc != jSrc)   ? w : 0.0f;

    // Contrastive: sim = g * invn_i * invn_j / temp(0.5)
    const float e    = __expf(g * invn[i] * inj * 2.0f);
    const bool  diag = (i == j);
    const bool  pos  = (!diag) && (lab[i] == lj) && (sub[i] != sj);
    float den = diag ? 0.0f : e;
    float num = pos  ? e    : 0.0f;
    float cnt = pos  ? 1.0f : 0.0f;

    // Column sums (mirrored contribution) accumulate raw per-element values.
    cden += den; cnum += num; ccnt += cnt;

    // Row reduce across the 16 lanes that hold the 16 columns of this row.
    #pragma unroll
    for (int o = 1; o < 16; o <<= 1) {
      den += __shfl_xor(den, o, 32);
      num += __shfl_xor(num, o, 32);
      cnt += __shfl_xor(cnt, o, 32);
    }
    if (m == 0) {
      atomicAdd(&rowDen[i], den);
      atomicAdd(&rowNum[i], num);
      atomicAdd(&rowCnt[i], cnt);
    }
  }

  if (offDiag) {
    // Mirrored block: column j of this tile is row j of G; combine the two
    // kh halves (rows i0..i0+7 and i0+8..i0+15) with one xor-16 shuffle.
    cden += __shfl_xor(cden, 16, 32);
    cnum += __shfl_xor(cnum, 16, 32);
    ccnt += __shfl_xor(ccnt, 16, 32);
    if (lane < 16) {
      atomicAdd(&rowDen[j], cden);
      atomicAdd(&rowNum[j], cnum);
      atomicAdd(&rowCnt[j], ccnt);
    }
  }

  // Wave-level reduction of MMD partials; off-diagonal tiles count twice.
  #pragma unroll
  for (int o = 1; o < 32; o <<= 1) {
    xx += __shfl_xor(xx, o, 32);
    yy += __shfl_xor(yy, o, 32);
    cr += __shfl_xor(cr, o, 32);
  }
  if (lane == 0) {
    const float wgt = offDiag ? 2.0f : 1.0f;
    atomicAdd(&scalars[0], wgt * xx);
    atomicAdd(&scalars[1], wgt * yy);
    atomicAdd(&scalars[2], wgt * cr);
  }
}

// Split-K WMMA GEMM: S_half = F_half^T * F_half (D x D), half = blockIdx.y.
__global__ void sil_coral_gemm(const float* __restrict__ F, float* __restrict__ S,
                               int B, int D, int kChunks) {
  const int lane = threadIdx.x & 31;
  const int wave = threadIdx.x >> 5;
  const int mid  = B >> 1;
  const int tilesPerDim = D >> 4;
  const int tilesTotal  = tilesPerDim * tilesPerDim;
  const int gw    = blockIdx.x * (blockDim.x >> 5) + wave;
  const int tile  = gw % tilesTotal;
  const int chunk = gw / tilesTotal;
  const int ti = tile % tilesPerDim;
  const int tj = tile / tilesPerDim;
  const int m  = lane & 15;
  const int kh = lane >> 4;

  const int di = ti * 16 + m;   // A: M index = feature dim of F^T
  const int dj = tj * 16 + m;   // B: N index
  const int kLen  = mid / kChunks;
  const int rBase = (blockIdx.y ? mid : 0) + chunk * kLen;
  const float* Fb = F + (size_t)rBase * D;

  v8f acc = {};
  for (int k = 0; k < kLen; k += 4) {
    const int kk = k + 2 * kh;  // K index = row of F within this half
    v2f a, b;
    a.x = Fb[(size_t)kk       * D + di];
    a.y = Fb[(size_t)(kk + 1) * D + di];
    b.x = Fb[(size_t)kk       * D + dj];
    b.y = Fb[(size_t)(kk + 1) * D + dj];
    acc = __builtin_amdgcn_wmma_f32_16x16x4_f32(false, a, false, b,
                                                (short)0, acc, false, false);
  }

  float* Sh = S + (size_t)blockIdx.y * D * D;
  #pragma unroll
  for (int r = 0; r < 8; ++r) {
    const int row = ti * 16 + r + 8 * kh;
    atomicAdd(&Sh[(size_t)row * D + tj * 16 + m], acc[r]);
  }
}

__global__ void sil_coral_fin(const float* __restrict__ S,
                              const float* __restrict__ meanS,
                              const float* __restrict__ meanT,
                              float* __restrict__ scalars, int B, int D) {
  __shared__ float sm[256];
  const int mid = B >> 1;
  const int total = D * D;
  const float inv = 1.0f / (float)(mid - 1);
  float part = 0.0f;
  for (int e = blockIdx.x * blockDim.x + threadIdx.x; e < total;
       e += gridDim.x * blockDim.x) {
    const int a = e / D, b = e % D;
    const float cs = (S[e]         - (float)mid * meanS[a] * meanS[b]) * inv;
    const float ct = (S[total + e] - (float)mid * meanT[a] * meanT[b]) * inv;
    const float d = cs - ct;
    part += d * d;
  }
  sm[threadIdx.x] = part; __syncthreads();
  for (int o = blockDim.x >> 1; o > 0; o >>= 1) {
    if (threadIdx.x < o) sm[threadIdx.x] += sm[threadIdx.x + o];
    __syncthreads();
  }
  if (threadIdx.x == 0) atomicAdd(&scalars[3], sm[0]);
}

__global__ void sil_contr_fin(const float* __restrict__ num,
                              const float* __restrict__ den,
                              const float* __restrict__ cnt,
                              float* __restrict__ scalars, int B) {
  __shared__ float sl[256];
  __shared__ float sv[256];
  const int i = blockIdx.x * blockDim.x + threadIdx.x;
  float loss = 0.0f, valid = 0.0f;
  if (i < B) {
    const float c = cnt[i];
    if (c > 0.0f) {
      const float numer = num[i] / fmaxf(c, 1.0f);
      loss  = -__logf(numer / (den[i] + 1e-8f));
      valid = 1.0f;
    }
  }
  sl[threadIdx.x] = loss; sv[threadIdx.x] = valid; __syncthreads();
  for (int o = blockDim.x >> 1; o > 0; o >>= 1) {
    if (threadIdx.x < o) {
      sl[threadIdx.x] += sl[threadIdx.x + o];
      sv[threadIdx.x] += sv[threadIdx.x + o];
    }
    __syncthreads();
  }
  if (threadIdx.x == 0) {
    atomicAdd(&scalars[4], sl[0]);
    atomicAdd(&scalars[5], sv[0]);
  }
}

__global__ void sil_combine(const float* __restrict__ scalars,
                            float* __restrict__ out, int B, int D) {
  const float mid = 0.5f * (float)B;
  // cross_sum counts both ordered directions => the -2*mean factor collapses
  // to -cross_sum/mid^2.
  const float mmd   = (scalars[0] + scalars[1] - scalars[2]) / (mid * mid);
  const float coral = scalars[3] / (4.0f * (float)D * (float)D);
  const float contr = scalars[4] / fmaxf(scalars[5], 1.0f);
  out[0] = mmd + coral + contr;
  out[1] = mmd;
  out[2] = coral;
  out[3] = contr;
}

extern "C" void kernel_launch(void* const* d_in, const int* in_sizes, int n_in,
                              void* d_out, int out_size, void* d_ws, size_t ws_size,
                              hipStream_t stream) {
  (void)n_in; (void)out_size; (void)ws_size;
  const float* F   = (const float*)d_in[0];
  const int*   lab = (const int*)d_in[1];
  const int*   sub = (const int*)d_in[2];
  const int B = in_sizes[1];          // 8192
  const int D = in_sizes[0] / B;      // 256

  float* ws    = (float*)d_ws;
  float* sq    = ws;
  float* invn  = sq   + B;
  float* num   = invn + B;
  float* den   = num  + B;
  float* cnt   = den  + B;
  float* meanS = cnt  + B;
  float* meanT = meanS + D;
  float* S     = meanT + D;           // 2*D*D
  float* scal  = S + 2 * D * D;       // 16
  const int wsFloats = 5 * B + 2 * D + 2 * D * D + 16;

  sil_init<<<(wsFloats + 255) / 256, 256, 0, stream>>>(ws, wsFloats);
  sil_row_norms<<<B / 8, 256, 0, stream>>>(F, sq, invn, D);
  sil_col_means<<<D, 256, 0, stream>>>(F, meanS, meanT, B, D);

  // Upper-triangle tiles only: T*(T+1)/2 waves, 8 waves per block.
  const long long T = B / 16;
  const long long totalTiles = T * (T + 1) / 2;     // 131328 for B=8192
  const int gramBlocks = (int)((totalTiles + 7) / 8);
  sil_gram_pass<<<gramBlocks, 256, 0, stream>>>(F, lab, sub, sq, invn,
                                                num, den, cnt, scal, B, D,
                                                totalTiles);

  const int kChunks = 8;
  const int wavesC  = (D / 16) * (D / 16) * kChunks;  // 2048 waves per half
  sil_coral_gemm<<<dim3(wavesC / 8, 2), 256, 0, stream>>>(F, S, B, D, kChunks);

  sil_coral_fin<<<64, 256, 0, stream>>>(S, meanS, meanT, scal, B, D);
  sil_contr_fin<<<(B + 255) / 256, 256, 0, stream>>>(num, den, cnt, scal, B);
  sil_combine<<<1, 1, 0, stream>>>(scal, (float*)d_out, B, D);
}